// Until_6236292514513
// MI455X (gfx1250) — compile-verified
//
#include <hip/hip_runtime.h>
#include <stdint.h>

// STL smooth "until", B=512, T=16384, a=0, b=127, tau=0.01.
// Transcendental-bound (3 v_exp/v_log per (t,k) step). Strategy:
//  - stage each 384-element window tile into LDS via CDNA5 async
//    global->LDS copies (ASYNCcnt path), s1/s2 interleaved as float2
//  - pre-scale once into base-2 log domain: v = (x/tau)*log2(e), so the
//    inner loop's exp/log are bare v_exp_f32/v_log_f32 (modifiers free)
//  - s2 tail padded with a huge negative so invalid k contribute exactly 0
//    to the running logsumexp (== reference's where(valid, ., -inf))

#define TILE      256
#define KWIN      128               // b + 1
#define LDS_ELEMS (TILE + KWIN)     // 384 >= TILE + halo(127)

__global__ __launch_bounds__(TILE) void until_kernel(
    const float* __restrict__ s1, const float* __restrict__ s2,
    float* __restrict__ out, int T)
{
    __shared__ float2 tile[LDS_ELEMS];

    const int tid = threadIdx.x;
    const int row = blockIdx.y;
    const int t0  = blockIdx.x * TILE;

    const float S        = 144.26950408889634f;      // (1/tau) * log2(e)
    const float OUTSCALE = 0.006931471805599453f;    // tau * ln(2)

    const float* __restrict__ s1r = s1 + (size_t)row * T;
    const float* __restrict__ s2r = s2 + (size_t)row * T;

    // Low 32 bits of a generic LDS pointer == LDS byte offset (aperture form).
    unsigned lds_base = (unsigned)(uintptr_t)&tile[0];

    // ---- Stage tile + halo via async global->LDS copies (raw values) ----
#pragma unroll
    for (int e = 0; e < 2; ++e) {
        int idx = tid + e * TILE;
        if (idx < LDS_ELEMS) {
            int g = t0 + idx;
            unsigned loff1 = lds_base + (unsigned)idx * 8u;
            unsigned loff2 = loff1 + 4u;
            if (g < T) {
                const float* g1 = s1r + g;
                const float* g2 = s2r + g;
                asm volatile("global_load_async_to_lds_b32 %0, %1, off"
                             :: "v"(loff1), "v"(g1) : "memory");
                asm volatile("global_load_async_to_lds_b32 %0, %1, off"
                             :: "v"(loff2), "v"(g2) : "memory");
            } else {
                // raw pads; scaled below with everything else:
                //   s1 -> +1e9  (window term vanishes, matches reference PAD_BIG)
                //   s2 -> -1e30 (candidate contributes 0 to M; finite after scale)
                tile[idx] = make_float2(1.0e9f, -1.0e30f);
            }
        }
    }
    asm volatile("s_wait_asynccnt 0" ::: "memory");
    __syncthreads();

    // ---- One-time scale into base-2 log domain (consumed 128x per thread) ----
#pragma unroll
    for (int e = 0; e < 2; ++e) {
        int idx = tid + e * TILE;
        if (idx < LDS_ELEMS) {
            float2 v = tile[idx];
            tile[idx] = make_float2(v.x * -S, v.y * S);   // x: -s1/tau*log2e, y: s2/tau*log2e
        }
    }
    __syncthreads();

    // Running log2-sum-exp2 in (max, sum) form; branchless rescale => 1 exp each.
    float mA = -INFINITY, sA = 0.0f;   // A2 = log2 sum 2^(-s1/tau*L2E) = mA + log2(sA)
    float mM = -INFINITY, sM = 0.0f;   // M2 = log2 sum 2^(cand/tau*L2E) = mM + log2(sM)

#pragma unroll 8
    for (int k = 0; k < KWIN; ++k) {
        float2 v = tile[tid + k];

        // A2 <- log2-add-exp2(A2, x)
        float x  = v.x;
        float d  = x - mA;
        float ed = __builtin_amdgcn_exp2f(-__builtin_fabsf(d)); // modifiers fold
        bool  gt = d > 0.0f;
        sA = gt ? __builtin_fmaf(sA, ed, 1.0f) : (sA + ed);
        mA = gt ? x : mA;

        // c2 = min(s2_scaled, -A2)
        float lg = __builtin_amdgcn_logf(sA);                   // v_log_f32 (log2)
        float c  = fminf(v.y, -(mA + lg));                      // neg folds into v_min

        // M2 <- log2-add-exp2(M2, c2)   (pads give 0-contribution, never max)
        float d2 = c - mM;
        float e2 = __builtin_amdgcn_exp2f(-__builtin_fabsf(d2));
        bool  gt2 = d2 > 0.0f;
        sM = gt2 ? __builtin_fmaf(sM, e2, 1.0f) : (sM + e2);
        mM = gt2 ? c : mM;
    }

    // out = tau * M_natural = tau*ln2 * (mM + log2(sM)); k=0 always valid => sM >= 1
    float res = OUTSCALE * (mM + __builtin_amdgcn_logf(sM));
    out[(size_t)row * T + t0 + tid] = res;
}

extern "C" void kernel_launch(void* const* d_in, const int* in_sizes, int n_in,
                              void* d_out, int out_size, void* d_ws, size_t ws_size,
                              hipStream_t stream) {
    (void)n_in; (void)out_size; (void)d_ws; (void)ws_size;
    const float* s1 = (const float*)d_in[0];
    const float* s2 = (const float*)d_in[1];
    float* out = (float*)d_out;

    const int T = 16384;
    const int B = in_sizes[0] / T;   // 512

    dim3 grid(T / TILE, B);
    until_kernel<<<grid, TILE, 0, stream>>>(s1, s2, out, T);
}